// FCOSHead_25022479466687
// MI455X (gfx1250) — compile-verified
//
#include <hip/hip_runtime.h>
#include <hip/hip_bf16.h>

#define NUM_CLASS 80
#define BATCH     16
#define NA_TOT    21824          // 16384+4096+1024+256+64
#define NSEL      3320           // 1000+1000+1000+256+64
#define NMSC      2000
#define MAXDET    100
#define SCORE_THR 0.05f
#define IOU_THR   0.5f
#define CLS_OFF   1025.0f        // IMG + 1

typedef float v2f __attribute__((ext_vector_type(2)));
typedef float v8f __attribute__((ext_vector_type(8)));

struct LvlPtrs {
  const float* cls[5];
  const float* bbox[5];
  const float* ctr[5];
};

__constant__ int d_hw[5]     = {128, 64, 32, 16, 8};
__constant__ int d_stridec[5]= {8, 16, 32, 64, 128};
__constant__ int d_n[5]      = {16384, 4096, 1024, 256, 64};

__device__ __forceinline__ float sigmoidf(float x) { return 1.0f / (1.0f + __expf(-x)); }

__device__ __forceinline__ int slot_level(int s) {
  return (s < 1000) ? 0 : (s < 2000) ? 1 : (s < 3000) ? 2 : (s < 3256) ? 3 : 4;
}

// ---------------------------------------------------------------------------
// Stage 1: per-anchor max class logit (bandwidth-bound streaming pass).
// float4 per thread, coalesced across the anchor dimension.
// ---------------------------------------------------------------------------
__global__ void kmax(const float* __restrict__ cls, float* __restrict__ maxlogit,
                     int n, int lvlOff) {
  const int groups = n >> 2;
  const int tid = blockIdx.x * blockDim.x + threadIdx.x;
  const int total = BATCH * groups;
  if (tid >= total) return;
  const int b = tid / groups, g = tid - b * groups;
  const float4* p = reinterpret_cast<const float4*>(cls) + (size_t)b * NUM_CLASS * groups + g;
  float4 m = p[0];
  for (int c = 1; c < NUM_CLASS; ++c) {
    const float4* q = p + (size_t)c * groups;
    __builtin_prefetch(q + groups, 0, 0);   // global_prefetch_b8: next class plane
    float4 v = *q;
    m.x = fmaxf(m.x, v.x); m.y = fmaxf(m.y, v.y);
    m.z = fmaxf(m.z, v.z); m.w = fmaxf(m.w, v.w);
  }
  float* o = maxlogit + (size_t)b * NA_TOT + lvlOff + 4 * g;
  o[0] = m.x; o[1] = m.y; o[2] = m.z; o[3] = m.w;
}

// ---------------------------------------------------------------------------
// Stage 2: nms_pre top-k per (image, level) via 2-pass radix select on
// ordered-float bits, then LDS-atomic compaction. One workgroup per image.
// ---------------------------------------------------------------------------
__global__ __launch_bounds__(1024) void ktopk(const float* __restrict__ maxlogit,
                                              int* __restrict__ sel_idx,
                                              int n, int k, int lvlOff, int koff) {
  const int b = blockIdx.x, t = threadIdx.x;
  const float* src = maxlogit + (size_t)b * NA_TOT + lvlOff;
  __shared__ unsigned hist[256];
  __shared__ int s_b1, s_above, s_thr, s_cnt;

  for (int i = t; i < 256; i += blockDim.x) hist[i] = 0;
  __syncthreads();
  for (int i = t; i < n; i += blockDim.x) {
    unsigned u = __float_as_uint(src[i]);
    u = (u & 0x80000000u) ? ~u : (u | 0x80000000u);
    atomicAdd(&hist[u >> 24], 1u);
  }
  __syncthreads();
  if (t == 0) {
    int cum = 0, b1 = 0, above = 0;
    for (int i = 255; i >= 0; --i) {
      int c = (int)hist[i];
      if (cum + c >= k) { b1 = i; above = cum; break; }
      cum += c;
    }
    s_b1 = b1; s_above = above; s_cnt = 0;
  }
  __syncthreads();
  const int b1 = s_b1;
  for (int i = t; i < 256; i += blockDim.x) hist[i] = 0;
  __syncthreads();
  for (int i = t; i < n; i += blockDim.x) {
    unsigned u = __float_as_uint(src[i]);
    u = (u & 0x80000000u) ? ~u : (u | 0x80000000u);
    if ((int)(u >> 24) == b1) atomicAdd(&hist[(u >> 16) & 0xFF], 1u);
  }
  __syncthreads();
  if (t == 0) {
    int cum = s_above, b2 = 0;
    for (int i = 255; i >= 0; --i) {
      int c = (int)hist[i];
      if (cum + c >= k) { b2 = i; break; }
      cum += c;
    }
    s_thr = (b1 << 8) | b2;
  }
  __syncthreads();
  const unsigned thr = (unsigned)s_thr;
  for (int i = t; i < n; i += blockDim.x) {
    unsigned u = __float_as_uint(src[i]);
    u = (u & 0x80000000u) ? ~u : (u | 0x80000000u);
    if ((u >> 16) >= thr) {
      int pos = atomicAdd(&s_cnt, 1);
      if (pos < k) sel_idx[b * NSEL + koff + pos] = i;
    }
  }
}

// ---------------------------------------------------------------------------
// Stage 3a: distance2bbox decode as a batched matmul on the WMMA unit.
// For 16 anchors per wave:  D(4x16) = M(4x8) . V(8x16), V col = [x,y,l,t,r,b,0,0],
// M rows: [1,0,-1,0,0,0] [0,1,0,-1,0,0] [1,0,0,0,1,0] [0,1,0,0,0,1].
// Two chained v_wmma_f32_16x16x4_f32 (K=0..3, K=4..7). Result lands in the C/D
// row-across-lanes layout: lanes 0..15, acc[0..3] = x1,y1,x2,y2 of slot=lane.
// ---------------------------------------------------------------------------
__device__ __forceinline__ float Mval(int r, int k) {
  if (r == 0) return (k == 0) ? 1.f : ((k == 2) ? -1.f : 0.f);
  if (r == 1) return (k == 1) ? 1.f : ((k == 3) ? -1.f : 0.f);
  if (r == 2) return (k == 0 || k == 4) ? 1.f : 0.f;
  if (r == 3) return (k == 1 || k == 5) ? 1.f : 0.f;
  return 0.f;
}

__global__ void kdecode(LvlPtrs P, const int* __restrict__ sel_idx,
                        const int* __restrict__ image_sizes,
                        float* __restrict__ sel_boxes) {
  const int b = blockIdx.y;
  const int wave = (blockIdx.x * blockDim.x + threadIdx.x) >> 5;
  const int lane = threadIdx.x & 31;
  const int half = lane >> 4;
  const int lane16 = lane & 15;
  const int slot = wave * 16 + lane16;
  const int slotc = (slot < NSEL) ? slot : (NSEL - 1);

  const int l = slot_level(slotc);
  const int a = sel_idx[b * NSEL + slotc];
  const int n_l = d_n[l], hwl = d_hw[l], sl = d_stridec[l];
  const int col = a % hwl, row = a / hwl;
  const float mx = (float)(col * sl + (sl >> 1));
  const float my = (float)(row * sl + (sl >> 1));
  const float* bb = P.bbox[l] + (size_t)b * 4 * n_l + a;
  const float dl = bb[0];
  const float dt = bb[(size_t)n_l];
  const float dr = bb[(size_t)2 * n_l];
  const float db = bb[(size_t)3 * n_l];

  // B operand (8x16, two K-slices of 4): lanes 0-15 carry K={0,1}/{4,5},
  // lanes 16-31 carry K={2,3}/{6,7} (mirrors the 32-bit A layout).
  v2f B0, B1, A0, A1;
  if (half == 0) { B0.x = mx; B0.y = my; B1.x = dr; B1.y = db; }
  else           { B0.x = dl; B0.y = dt; B1.x = 0.f; B1.y = 0.f; }
  A0.x = Mval(lane16, half ? 2 : 0); A0.y = Mval(lane16, half ? 3 : 1);
  A1.x = Mval(lane16, half ? 6 : 4); A1.y = Mval(lane16, half ? 7 : 5);

  v8f acc = {};
  acc = __builtin_amdgcn_wmma_f32_16x16x4_f32(false, A0, false, B0, (short)0, acc, false, false);
  acc = __builtin_amdgcn_wmma_f32_16x16x4_f32(false, A1, false, B1, (short)0, acc, false, false);

  if (half == 0 && slot < NSEL) {
    const float h = (float)image_sizes[b * 2 + 0];
    const float w = (float)image_sizes[b * 2 + 1];
    float x1 = fminf(fmaxf(acc[0], 0.f), w);
    float y1 = fminf(fmaxf(acc[1], 0.f), h);
    float x2 = fminf(fmaxf(acc[2], 0.f), w);
    float y2 = fminf(fmaxf(acc[3], 0.f), h);
    float* o = sel_boxes + ((size_t)b * NSEL + slot) * 4;
    o[0] = x1; o[1] = y1; o[2] = x2; o[3] = y2;
  }
}

// ---------------------------------------------------------------------------
// Stage 3b: gather selected anchors, final = sigmoid(cls)*sigmoid(ctr), thresh.
// ---------------------------------------------------------------------------
__global__ void kscores(LvlPtrs P, const int* __restrict__ sel_idx,
                        float* __restrict__ sel_scores) {
  const int slot = blockIdx.x * blockDim.x + threadIdx.x;
  const int b = blockIdx.y;
  if (slot >= NSEL) return;
  const int l = slot_level(slot);
  const int a = sel_idx[b * NSEL + slot];
  const int n_l = d_n[l];
  const float cs = sigmoidf(P.ctr[l][(size_t)b * n_l + a]);
  const float* cb = P.cls[l] + (size_t)b * NUM_CLASS * n_l + a;
  float* o = sel_scores + ((size_t)b * NSEL + slot) * NUM_CLASS;
  for (int c = 0; c < NUM_CLASS; ++c) {
    float s = sigmoidf(cb[(size_t)c * n_l]) * cs;
    o[c] = (s > SCORE_THR) ? s : 0.f;
  }
}

// ---------------------------------------------------------------------------
// Stage 4: per-image top-2000 of (NSEL x 80) positives via radix select.
// ---------------------------------------------------------------------------
__global__ __launch_bounds__(1024) void ktop2000(const float* __restrict__ sel_scores,
                                                 const float* __restrict__ sel_boxes,
                                                 float* __restrict__ cand_score,
                                                 float* __restrict__ cand_box,
                                                 int* __restrict__ cand_label) {
  const int b = blockIdx.x, t = threadIdx.x;
  const float* src = sel_scores + (size_t)b * NSEL * NUM_CLASS;
  const int n = NSEL * NUM_CLASS;
  __shared__ unsigned hist[256];
  __shared__ int s_b1, s_above, s_thr, s_all, s_cnt;

  for (int i = t; i < NMSC; i += blockDim.x) {
    cand_score[b * NMSC + i] = 0.f;
    cand_label[b * NMSC + i] = -1;
    float* cb = cand_box + ((size_t)b * NMSC + i) * 4;
    cb[0] = cb[1] = cb[2] = cb[3] = 0.f;
  }
  for (int i = t; i < 256; i += blockDim.x) hist[i] = 0;
  __syncthreads();
  for (int i = t; i < n; i += blockDim.x) {
    float v = src[i];
    if (v > 0.f) atomicAdd(&hist[(__float_as_uint(v) >> 23) & 0xFF], 1u);
  }
  __syncthreads();
  if (t == 0) {
    int cum = 0, b1 = -1, above = 0;
    for (int i = 255; i >= 0; --i) {
      int c = (int)hist[i];
      if (cum + c >= NMSC) { b1 = i; above = cum; break; }
      cum += c;
    }
    s_all = (b1 < 0);          // fewer than NMSC positives -> take all
    s_b1 = b1; s_above = above; s_cnt = 0;
  }
  __syncthreads();
  const int all = s_all, b1 = s_b1;
  if (!all) {
    for (int i = t; i < 256; i += blockDim.x) hist[i] = 0;
    __syncthreads();
    for (int i = t; i < n; i += blockDim.x) {
      float v = src[i];
      if (v > 0.f) {
        unsigned u = __float_as_uint(v);
        if (((u >> 23) & 0xFF) == (unsigned)b1) atomicAdd(&hist[(u >> 15) & 0xFF], 1u);
      }
    }
    __syncthreads();
    if (t == 0) {
      int cum = s_above, b2 = 0;
      for (int i = 255; i >= 0; --i) {
        int c = (int)hist[i];
        if (cum + c >= NMSC) { b2 = i; break; }
        cum += c;
      }
      s_thr = (b1 << 8) | b2;
    }
    __syncthreads();
  }
  const int thr = all ? 0 : s_thr;
  __syncthreads();
  for (int i = t; i < n; i += blockDim.x) {
    float v = src[i];
    if (v > 0.f && (int)(__float_as_uint(v) >> 15) >= thr) {
      int pos = atomicAdd(&s_cnt, 1);
      if (pos < NMSC) {
        int slot = i / NUM_CLASS, c = i - slot * NUM_CLASS;
        cand_score[b * NMSC + pos] = v;
        cand_label[b * NMSC + pos] = c;
        const float* sb = sel_boxes + ((size_t)b * NSEL + slot) * 4;
        float* cb = cand_box + ((size_t)b * NMSC + pos) * 4;
        cb[0] = sb[0]; cb[1] = sb[1]; cb[2] = sb[2]; cb[3] = sb[3];
      }
    }
  }
}

// ---------------------------------------------------------------------------
// Stage 5: sequential NMS (100 argmax+suppress steps), all state in LDS.
// One workgroup per image; 16 images in flight across WGPs.
// ---------------------------------------------------------------------------
__global__ __launch_bounds__(1024) void knms(const float* __restrict__ cand_score,
                                             const float* __restrict__ cand_box,
                                             const int* __restrict__ cand_label,
                                             float* __restrict__ out) {
  const int b = blockIdx.x, t = threadIdx.x;
  __shared__ float ssc[NMSC];
  __shared__ float sbx[NMSC * 4];   // class-offset boxes
  __shared__ int   slb[NMSC];
  __shared__ float rval[1024];
  __shared__ int   ridx[1024];

  for (int i = t; i < NMSC; i += blockDim.x) {
    float sc = cand_score[b * NMSC + i];
    int lb = cand_label[b * NMSC + i];
    float off = (lb < 0) ? 0.f : (float)lb * CLS_OFF;
    const float* cb = cand_box + ((size_t)b * NMSC + i) * 4;
    ssc[i] = sc;
    sbx[i * 4 + 0] = cb[0] + off; sbx[i * 4 + 1] = cb[1] + off;
    sbx[i * 4 + 2] = cb[2] + off; sbx[i * 4 + 3] = cb[3] + off;
    slb[i] = lb;
  }
  __syncthreads();

  float* dets = out;                         // (B, 100, 5)
  float* labs = out + (size_t)BATCH * MAXDET * 5;  // (B, 100) as float

  for (int it = 0; it < MAXDET; ++it) {
    float bv = -1.f; int bi = 0;
    for (int i = t; i < NMSC; i += blockDim.x) {
      float v = ssc[i];
      if (v > bv) { bv = v; bi = i; }
    }
    rval[t] = bv; ridx[t] = bi;
    __syncthreads();
    for (int s = 512; s > 0; s >>= 1) {
      if (t < s && rval[t + s] > rval[t]) { rval[t] = rval[t + s]; ridx[t] = ridx[t + s]; }
      __syncthreads();
    }
    const int j = ridx[0];
    const float wv = rval[0];
    const float wx1 = sbx[j * 4 + 0], wy1 = sbx[j * 4 + 1];
    const float wx2 = sbx[j * 4 + 2], wy2 = sbx[j * 4 + 3];
    __syncthreads();

    if (t == 0) {
      bool valid = wv > 0.f;
      float off = (slb[j] >= 0) ? (float)slb[j] * CLS_OFF : 0.f;
      float* dr = dets + ((size_t)b * MAXDET + it) * 5;
      dr[0] = valid ? (wx1 - off) : 0.f;
      dr[1] = valid ? (wy1 - off) : 0.f;
      dr[2] = valid ? (wx2 - off) : 0.f;
      dr[3] = valid ? (wy2 - off) : 0.f;
      dr[4] = valid ? wv : 0.f;
      labs[b * MAXDET + it] = valid ? (float)slb[j] : -1.f;
    }
    const float wa = (wx2 - wx1) * (wy2 - wy1);
    for (int i = t; i < NMSC; i += blockDim.x) {
      float x1 = fmaxf(wx1, sbx[i * 4 + 0]), y1 = fmaxf(wy1, sbx[i * 4 + 1]);
      float x2 = fminf(wx2, sbx[i * 4 + 2]), y2 = fminf(wy2, sbx[i * 4 + 3]);
      float iw = fmaxf(x2 - x1, 0.f), ih = fmaxf(y2 - y1, 0.f);
      float inter = iw * ih;
      float a2 = (sbx[i * 4 + 2] - sbx[i * 4 + 0]) * (sbx[i * 4 + 3] - sbx[i * 4 + 1]);
      float iou = inter / fmaxf(wa + a2 - inter, 1e-6f);
      if (iou >= IOU_THR) ssc[i] = 0.f;
    }
    __syncthreads();
  }
}

// ---------------------------------------------------------------------------
extern "C" void kernel_launch(void* const* d_in, const int* in_sizes, int n_in,
                              void* d_out, int out_size, void* d_ws, size_t ws_size,
                              hipStream_t stream) {
  (void)in_sizes; (void)n_in; (void)out_size; (void)ws_size;
  static const int NL[5]   = {16384, 4096, 1024, 256, 64};
  static const int LOFF[5] = {0, 16384, 20480, 21504, 21760};
  static const int KS[5]   = {1000, 1000, 1000, 256, 64};
  static const int KOFF[5] = {0, 1000, 2000, 3000, 3256};

  LvlPtrs P;
  for (int l = 0; l < 5; ++l) {
    P.cls[l]  = (const float*)d_in[3 * l + 0];
    P.bbox[l] = (const float*)d_in[3 * l + 1];
    P.ctr[l]  = (const float*)d_in[3 * l + 2];
  }
  const int* imsz = (const int*)d_in[15];

  char* w = (char*)d_ws;
  size_t off = 0;
  auto take = [&](size_t bytes) -> void* {
    void* p = w + off;
    off = (off + bytes + 255) & ~(size_t)255;
    return p;
  };
  float* maxlogit   = (float*)take((size_t)BATCH * NA_TOT * 4);
  int*   sel_idx    = (int*)  take((size_t)BATCH * NSEL * 4);
  float* sel_scores = (float*)take((size_t)BATCH * NSEL * NUM_CLASS * 4);
  float* sel_boxes  = (float*)take((size_t)BATCH * NSEL * 4 * 4);
  float* cand_score = (float*)take((size_t)BATCH * NMSC * 4);
  float* cand_box   = (float*)take((size_t)BATCH * NMSC * 4 * 4);
  int*   cand_label = (int*)  take((size_t)BATCH * NMSC * 4);

  for (int l = 0; l < 5; ++l) {
    int total = BATCH * (NL[l] >> 2);
    kmax<<<(total + 255) / 256, 256, 0, stream>>>(P.cls[l], maxlogit, NL[l], LOFF[l]);
  }
  for (int l = 0; l < 5; ++l) {
    ktopk<<<BATCH, 1024, 0, stream>>>(maxlogit, sel_idx, NL[l], KS[l], LOFF[l], KOFF[l]);
  }
  {
    dim3 g(26, BATCH);  // 26 blocks * 8 waves = 208 tiles of 16 slots >= 3320
    kdecode<<<g, 256, 0, stream>>>(P, sel_idx, imsz, sel_boxes);
  }
  {
    dim3 g((NSEL + 255) / 256, BATCH);
    kscores<<<g, 256, 0, stream>>>(P, sel_idx, sel_scores);
  }
  ktop2000<<<BATCH, 1024, 0, stream>>>(sel_scores, sel_boxes, cand_score, cand_box, cand_label);
  knms<<<BATCH, 1024, 0, stream>>>(cand_score, cand_box, cand_label, (float*)d_out);
}